// ToRGBLayer_69020124447371
// MI455X (gfx1250) — compile-verified
//
#include <hip/hip_runtime.h>

typedef float v2f __attribute__((ext_vector_type(2)));
typedef float v4f __attribute__((ext_vector_type(4)));
typedef float v8f __attribute__((ext_vector_type(8)));

#define NIMG 8
#define HW   16384      // 128*128 pixels per image
#define CIN  512
#define FOUT 3
#define FPAD 16
#define LDSS 516        // 512 + 4 pad: b128 reads hit distinct banks

// ---------------------------------------------------------------------------
// Kernel 1: per-image effective weight  weff[img][f][c] = s[img][c]*cw[c][f]
//   s = dlatents[:,17] @ (affine_w / sqrt(512)) + affine_b
//   cw scaled by 1/sqrt(1*1*512). F padded to 16 with zeros so kernel 2's
//   WMMA B-matrix columns 3..15 contribute nothing.
// ---------------------------------------------------------------------------
__global__ __launch_bounds__(256) void torgb_affine_kernel(
    const float* __restrict__ dlatents, const float* __restrict__ affine_w,
    const float* __restrict__ affine_b, const float* __restrict__ conv_w,
    float* __restrict__ wst_g)
{
    __shared__ float dl[CIN];
    const int img = blockIdx.x;
    const int t   = threadIdx.x;

    const float* d = dlatents + ((size_t)img * 18 + 17) * CIN;
    for (int i = t; i < CIN; i += 256) dl[i] = d[i];
    __syncthreads();

    const float ascale = 0.044194173824159216f;  // 1/sqrt(512)
    const float cscale = 0.044194173824159216f;  // 1/sqrt(1*1*512)

    const int c0 = 2 * t;                        // each thread: 2 channels
    float s0 = 0.f, s1 = 0.f;
    for (int w = 0; w < CIN; ++w) {
        v2f aw = *(const v2f*)(affine_w + (size_t)w * CIN + c0);
        float dv = dl[w];
        s0 = fmaf(dv, aw.x, s0);
        s1 = fmaf(dv, aw.y, s1);
    }
    s0 = s0 * ascale + affine_b[c0];
    s1 = s1 * ascale + affine_b[c0 + 1];

    float* wimg = wst_g + (size_t)img * (FPAD * CIN);
    for (int f = 0; f < FPAD; ++f) {
        float w0 = 0.f, w1 = 0.f;
        if (f < FOUT) {
            w0 = s0 * conv_w[c0 * FOUT + f] * cscale;
            w1 = s1 * conv_w[(c0 + 1) * FOUT + f] * cscale;
        }
        wimg[f * CIN + c0]     = w0;
        wimg[f * CIN + c0 + 1] = w1;
    }
}

// ---------------------------------------------------------------------------
// Kernel 2: streaming 1x1 conv via V_WMMA_F32_16X16X4_F32.
//   Per wave: two M=16 pixel tiles (independent WMMA chains), N = 16
//   features (3 live), K = 512 channels. Channel->k-slot permutation:
//   8-channel chunk per float4 load, half-wave h covers channels
//   cb+4h..cb+4h+3 feeding two WMMAs per tile; one ds_load_b128 feeds
//   all four WMMAs of the chunk.
// ---------------------------------------------------------------------------
__global__ __launch_bounds__(256) void torgb_conv_kernel(
    const float* __restrict__ x, const float* __restrict__ y,
    const float* __restrict__ wst_g, const float* __restrict__ conv_b,
    float* __restrict__ out)
{
    __shared__ float wl[FPAD * LDSS];

    const int img      = blockIdx.x >> 6;     // 64 blocks per image
    const int blkInImg = blockIdx.x & 63;
    const int t        = threadIdx.x;

    // Stage weff[img] into LDS, transposed-padded [f][c] with stride 516.
    const float* wg = wst_g + (size_t)img * (FPAD * CIN);
    for (int i = t; i < FPAD * CIN; i += 256) {
        int f = i >> 9;          // /512
        int c = i & 511;
        wl[f * LDSS + c] = wg[i];
    }
    __syncthreads();

    const int wave = t >> 5;
    const int lane = t & 31;
    const int mn   = lane & 15;  // A: pixel-in-tile; B/D: feature column
    const int h    = lane >> 4;  // half-wave -> k-slot group

    const int p0 = blkInImg * 256 + wave * 32;           // 2 tiles / wave
    const float* xr0 = x + ((size_t)img * HW + p0 + mn) * CIN;
    const float* xr1 = xr0 + (size_t)16 * CIN;
    const float* wr  = wl + mn * LDSS;

    v8f acc0 = {};
    v8f acc1 = {};
    for (int cb = 0; cb < CIN; cb += 8) {
        if ((cb & 31) == 0) {                            // once per 128B line
            __builtin_prefetch(xr0 + cb + 256, 0, 0);    // ~1KB ahead
            __builtin_prefetch(xr1 + cb + 256, 0, 0);
        }
        v4f xa0 = __builtin_nontemporal_load((const v4f*)(xr0 + cb + 4 * h));
        v4f xa1 = __builtin_nontemporal_load((const v4f*)(xr1 + cb + 4 * h));
        v4f wb  = *(const v4f*)(wr + cb + 4 * h);        // ds_load_b128
        v2f b1 = {wb.x, wb.y}, b2 = {wb.z, wb.w};
        v2f a01 = {xa0.x, xa0.y}, a02 = {xa0.z, xa0.w};
        v2f a11 = {xa1.x, xa1.y}, a12 = {xa1.z, xa1.w};
        acc0 = __builtin_amdgcn_wmma_f32_16x16x4_f32(
                   false, a01, false, b1, (short)0, acc0, false, false);
        acc1 = __builtin_amdgcn_wmma_f32_16x16x4_f32(
                   false, a11, false, b1, (short)0, acc1, false, false);
        acc0 = __builtin_amdgcn_wmma_f32_16x16x4_f32(
                   false, a02, false, b2, (short)0, acc0, false, false);
        acc1 = __builtin_amdgcn_wmma_f32_16x16x4_f32(
                   false, a12, false, b2, (short)0, acc1, false, false);
    }

    // Epilogue: D[m][n] lives in acc[j] with m = j + 8h, n = mn (cols 0..2).
    if (mn < FOUT) {
        const float bias = conv_b[mn];                   // LR_MULT == 1
        for (int tt = 0; tt < 2; ++tt) {
            v8f acc = tt ? acc1 : acc0;
            for (int j = 0; j < 8; ++j) {
                int p = p0 + 16 * tt + j + 8 * h;
                size_t oidx = ((size_t)img * HW + p) * FOUT + mn;
                float v = acc[j] + bias;
                v = fminf(fmaxf(v, -256.f), 256.f);
                v += __builtin_nontemporal_load(y + oidx);
                __builtin_nontemporal_store(v, out + oidx);
            }
        }
    }
}

// ---------------------------------------------------------------------------
extern "C" void kernel_launch(void* const* d_in, const int* in_sizes, int n_in,
                              void* d_out, int out_size, void* d_ws, size_t ws_size,
                              hipStream_t stream)
{
    (void)in_sizes; (void)n_in; (void)out_size; (void)ws_size;
    const float* x        = (const float*)d_in[0];
    const float* y        = (const float*)d_in[1];
    const float* dlatents = (const float*)d_in[2];
    const float* affine_w = (const float*)d_in[3];
    const float* affine_b = (const float*)d_in[4];
    const float* conv_w   = (const float*)d_in[5];
    const float* conv_b   = (const float*)d_in[6];
    float* out = (float*)d_out;
    float* wst = (float*)d_ws;   // 8*16*512 floats = 256 KB scratch

    torgb_affine_kernel<<<NIMG, 256, 0, stream>>>(dlatents, affine_w, affine_b,
                                                  conv_w, wst);
    torgb_conv_kernel<<<NIMG * 64, 256, 0, stream>>>(x, y, wst, conv_b, out);
}